// Proposal_90933047591045
// MI455X (gfx1250) — compile-verified
//
#include <hip/hip_runtime.h>
#include <hip/hip_bf16.h>

// ---------------------------------------------------------------------------
// Problem constants (from the reference)
// ---------------------------------------------------------------------------
#define NB    16
#define NH    512
#define NW    512
#define HS    54
#define PW    51          // HS - WIN + 1
#define CROPN 64
#define NCH_AB 6
#define NCH_B  3

#define AB_ELEMS   (NB * NCH_AB * NH * NW)          // 25,165,824
#define CROP_ELEMS (NB * NCH_B * CROPN * CROPN)     // 196,608

// ---------------------------------------------------------------------------
// CDNA5 async global<->LDS data mover (gfx1250). The clang builtins take
// v4i32 pointers in addrspace(1)/(3) for the b128 variants (per compiler
// diagnostics), so cast to those exact types.
// ---------------------------------------------------------------------------
#if defined(__gfx1250__) && \
    __has_builtin(__builtin_amdgcn_global_load_async_to_lds_b128) && \
    __has_builtin(__builtin_amdgcn_global_store_async_from_lds_b128)
#define HAVE_ASYNC_LDS 1
typedef int v4i_t __attribute__((ext_vector_type(4)));
typedef __attribute__((address_space(1))) v4i_t gas_v4i;
typedef __attribute__((address_space(3))) v4i_t las_v4i;
#else
#define HAVE_ASYNC_LDS 0
#endif

#if __has_builtin(__builtin_amdgcn_s_wait_asynccnt)
#define WAIT_ASYNC0() __builtin_amdgcn_s_wait_asynccnt(0)
#else
#define WAIT_ASYNC0() asm volatile("s_wait_asynccnt 0" ::: "memory")
#endif

// ---------------------------------------------------------------------------
// Kernel 1: per-batch argmax over 51x51 crop of the score map; emit box
// floats + integer patch coords into the workspace (8 dwords per batch).
// ws layout per batch: [0..3] = ny0, nx0, ny1, nx1 (f32); [4] = xi; [5] = yi.
// ---------------------------------------------------------------------------
__global__ void localize_kernel(const float* __restrict__ score,
                                float* __restrict__ wsf) {
    const int b = blockIdx.x;
    const float* m = score + (size_t)b * HS * HS;   // channel dim is 1

    __shared__ float sval[256];
    __shared__ int   sidx[256];

    float best = -INFINITY;
    int   bidx = 0x7FFFFFFF;
    for (int t = threadIdx.x; t < PW * PW; t += blockDim.x) {
        const int row = t / PW, col = t % PW;
        const float v = m[row * HS + col];
        if (v > best) { best = v; bidx = t; }       // ascending t => first max
    }
    sval[threadIdx.x] = best;
    sidx[threadIdx.x] = bidx;
    __syncthreads();

    for (int s = 128; s > 0; s >>= 1) {
        if ((int)threadIdx.x < s) {
            const float v2 = sval[threadIdx.x + s];
            const int   i2 = sidx[threadIdx.x + s];
            const float v1 = sval[threadIdx.x];
            const int   i1 = sidx[threadIdx.x];
            if (v2 > v1 || (v2 == v1 && i2 < i1)) { // global first-occurrence
                sval[threadIdx.x] = v2;
                sidx[threadIdx.x] = i2;
            }
        }
        __syncthreads();
    }

    if (threadIdx.x == 0) {
        const float bestv = sval[0];
        const int   idx   = sidx[0];
        const int x = (bestv > 0.0f) ? (idx / PW) : 0;
        const int y = (bestv > 0.0f) ? (idx % PW) : 0;

        const float stride = floorf((512.0f - 70.0f) / 54.0f);  // 8.0
        const float ax0 = (float)x * stride + 70.0f;            // row coord
        const float ax1 = (float)y * stride + 70.0f;            // col coord

        // _boxes: x1 = ax[:,1], y1 = ax[:,0]; sw = sh = 1 exactly.
        const float x1 = ax1, y1 = ax0;
        const float nx0 = (x1 + 0.5f - 0.5f) / (float)(NW - 1);
        const float ny0 = (y1 + 0.5f - 0.5f) / (float)(NH - 1);
        const float nw  = 1.0f * (float)(CROPN - 1) / (float)(NW - 1);
        const float nh  = 1.0f * (float)(CROPN - 1) / (float)(NH - 1);

        float* wb = wsf + b * 8;
        wb[0] = ny0;
        wb[1] = nx0;
        wb[2] = ny0 + nh;
        wb[3] = nx0 + nw;
        ((int*)wb)[4] = (int)ax0;   // xi (H index of patch)
        ((int*)wb)[5] = (int)ax1;   // yi (W index of patch)
    }
}

// ---------------------------------------------------------------------------
// Kernel 2: 100 MB bulk copy real_AB -> out via the CDNA5 async data mover.
// Each lane stages 4x16B chunks through block-private LDS per iteration;
// waits are per-wave (lanes only touch their own LDS slots -> no barrier).
// ---------------------------------------------------------------------------
__global__ void __launch_bounds__(256)
bulk_copy_kernel(const float4* __restrict__ src, float4* __restrict__ dst,
                 int nChunks) {
#if HAVE_ASYNC_LDS
    __shared__ float4 buf[4 * 256];     // 16 KB staging
    const int t = threadIdx.x;
    for (int base = blockIdx.x * 1024; base < nChunks;
         base += gridDim.x * 1024) {
#pragma unroll
        for (int k = 0; k < 4; ++k) {
            const int idx = base + (k << 8) + t;
            if (idx < nChunks) {
                __builtin_amdgcn_global_load_async_to_lds_b128(
                    (gas_v4i*)(src + idx),
                    (las_v4i*)(buf + (k << 8) + t), 0, 0);
            }
        }
        WAIT_ASYNC0();                   // loads landed in LDS
#pragma unroll
        for (int k = 0; k < 4; ++k) {
            const int idx = base + (k << 8) + t;
            if (idx < nChunks) {
                __builtin_amdgcn_global_store_async_from_lds_b128(
                    (gas_v4i*)(dst + idx),
                    (las_v4i*)(buf + (k << 8) + t), 0, 0);
            }
        }
        WAIT_ASYNC0();                   // LDS free for next iteration
    }
#else
    const int stride = gridDim.x * blockDim.x;
    for (int i = blockIdx.x * blockDim.x + threadIdx.x; i < nChunks;
         i += stride)
        dst[i] = src[i];
#endif
}

// ---------------------------------------------------------------------------
// Kernel 3: overwrite the 96 patched elements: out[b, c, xi, yi] = fake_AB[...]
// ---------------------------------------------------------------------------
__global__ void patch_kernel(const float* __restrict__ fake_AB,
                             const float* __restrict__ wsf,
                             float* __restrict__ out) {
    const int g = threadIdx.x;
    if (g >= NB * NCH_AB) return;
    const int b = g / NCH_AB, c = g % NCH_AB;
    const int xi = ((const int*)(wsf + b * 8))[4];
    const int yi = ((const int*)(wsf + b * 8))[5];
    const size_t off = (((size_t)b * NCH_AB + c) * NH + xi) * NW + yi;
    out[off] = fake_AB[off];
}

// ---------------------------------------------------------------------------
// Kernel 4: crop-and-resize (bilinear, replicating the reference float math)
// for both real_B and fake_B. which = gid / CROP_ELEMS selects the image.
// ---------------------------------------------------------------------------
__global__ void crop_kernel(const float* __restrict__ real_B,
                            const float* __restrict__ fake_B,
                            const float* __restrict__ wsf,
                            float* __restrict__ out_real,
                            float* __restrict__ out_fake) {
    const int gid = blockIdx.x * blockDim.x + threadIdx.x;
    if (gid >= 2 * CROP_ELEMS) return;
    const int which = gid / CROP_ELEMS;          // 0 = real, 1 = fake
    int r = gid % CROP_ELEMS;
    const int j = r % CROPN;  r /= CROPN;
    const int i = r % CROPN;  r /= CROPN;
    const int c = r % NCH_B;
    const int b = r / NCH_B;

    const float* wb = wsf + b * 8;
    const float ny0 = wb[0], nx0 = wb[1], ny1 = wb[2], nx1 = wb[3];

    // ys = ny0*(H-1) + i*(ny1-ny0)*(H-1)/(CROP-1)   (same op order as ref)
    float ty = (float)i * (ny1 - ny0); ty = ty * 511.0f; ty = ty / 63.0f;
    const float ys = ny0 * 511.0f + ty;
    float tx = (float)j * (nx1 - nx0); tx = tx * 511.0f; tx = tx / 63.0f;
    const float xs = nx0 * 511.0f + tx;

    const bool vy = (ys >= 0.0f) && (ys <= 511.0f);
    const bool vx = (xs >= 0.0f) && (xs <= 511.0f);

    const float y0f = floorf(ys), x0f = floorf(xs);
    const int y0 = (int)fminf(fmaxf(y0f, 0.0f), 511.0f);
    const int x0 = (int)fminf(fmaxf(x0f, 0.0f), 511.0f);
    const int y1 = min(y0 + 1, 511);
    const int x1 = min(x0 + 1, 511);
    const float ly = ys - y0f;
    const float lx = xs - x0f;

    const float* img = (which ? fake_B : real_B) +
                       (((size_t)b * NCH_B + c) * NH) * NW;
    const float a00 = img[(size_t)y0 * NW + x0];
    const float a01 = img[(size_t)y0 * NW + x1];
    const float a10 = img[(size_t)y1 * NW + x0];
    const float a11 = img[(size_t)y1 * NW + x1];

    const float top = (1.0f - lx) * a00 + lx * a01;
    const float bot = (1.0f - lx) * a10 + lx * a11;
    float o = (1.0f - ly) * top + ly * bot;
    o = (vy && vx) ? o : 0.0f;

    float* outp = which ? out_fake : out_real;
    outp[(((size_t)b * NCH_B + c) * CROPN + i) * CROPN + j] = o;
}

// ---------------------------------------------------------------------------
// Launch
// ---------------------------------------------------------------------------
extern "C" void kernel_launch(void* const* d_in, const int* in_sizes, int n_in,
                              void* d_out, int out_size, void* d_ws,
                              size_t ws_size, hipStream_t stream) {
    const float* real_AB  = (const float*)d_in[0];
    const float* fake_AB  = (const float*)d_in[1];
    const float* score    = (const float*)d_in[2];
    const float* real_B   = (const float*)d_in[3];
    const float* fake_B   = (const float*)d_in[4];

    float* out      = (float*)d_out;
    float* out_ABm  = out;                                // 25,165,824
    float* out_realBr = out + (size_t)AB_ELEMS;           // 196,608
    float* out_fakeBr = out + (size_t)AB_ELEMS + CROP_ELEMS;

    float* wsf = (float*)d_ws;                            // 16 * 8 dwords

    // 1) localize + box precompute (tiny)
    localize_kernel<<<NB, 256, 0, stream>>>(score, wsf);

    // 2) 100 MB streaming copy via async global<->LDS (HBM-bound, ~4.3 us
    //    each direction at 23.3 TB/s)
    const int nChunks = AB_ELEMS / 4;                     // float4 chunks
    bulk_copy_kernel<<<2048, 256, 0, stream>>>(
        (const float4*)real_AB, (float4*)out_ABm, nChunks);

    // 3) patch the 96 selected elements (must follow the bulk copy)
    patch_kernel<<<1, 128, 0, stream>>>(fake_AB, wsf, out_ABm);

    // 4) both bilinear crops
    crop_kernel<<<(2 * CROP_ELEMS + 255) / 256, 256, 0, stream>>>(
        real_B, fake_B, wsf, out_realBr, out_fakeBr);
}